// HardL1ACELoss_83322365542591
// MI455X (gfx1250) — compile-verified
//
#include <hip/hip_runtime.h>
#include <stdint.h>

// ---------------------------------------------------------------------------
// HardL1ACELoss (adaptive calibration error) for MI455X / gfx1250.
//
// Memory-bound: 256 MB read @ 23.3 TB/s => ~11 us floor. Strategy:
//  * stream preds/targets global->LDS with the CDNA5 async-load engine
//    (GLOBAL_LOAD_ASYNC_TO_LDS_B128 / ASYNCcnt), double buffered,
//    2048-element tiles per 256-thread block, non-temporal (one-shot stream
//    larger than the 192MB L2)
//  * per-half-wave LDS histogram replicas (16 per block) to minimize
//    same-bin LDS atomic serialization; 2 integer LDS atomics per element
//  * all-integer accumulation (fixed point 2^16 for sum_p) => bitwise
//    deterministic across graph replays (no float atomics anywhere)
// ---------------------------------------------------------------------------

#define N_BINS 20
#define TILE   2048              // elements per block tile
#define TPB    256               // 8 x wave32
#define NREP   16                // one histogram replica per half-wave
#define HSTRIDE 64               // u32 words per replica (spans all 64 banks)
#define MAX_GRID 4096
#define MAX_TILES_PER_BLOCK 32   // keeps 16-bit packed counters from overflowing

__device__ __forceinline__ unsigned lds_off(const void* p) {
    // CDNA5 flat LDS aperture: addr[31:0] is the wave-relative LDS byte offset.
    return (unsigned)(uintptr_t)p;
}

__device__ __forceinline__ void async_cp16(unsigned lds, unsigned long long gaddr) {
    // GLOBAL_LOAD_ASYNC_TO_LDS_B128: vdst = per-lane LDS byte address,
    // vaddr = per-lane 64-bit global address, SADDR=NULL (GV mode).
    // Non-temporal: single-pass 256MB stream > 192MB L2, don't thrash it.
    asm volatile("global_load_async_to_lds_b128 %0, %1, off th:TH_LOAD_NT"
                 :: "v"(lds), "v"(gaddr)
                 : "memory");
}

__device__ __forceinline__ void acc_elem(float p, int t, unsigned* whist) {
    if (p >= 0.0f && p < 1.0f) {
        int b = (int)(p * (float)N_BINS);
        if (b > N_BINS - 1) b = N_BINS - 1;
        // packed: cnt in [31:16], sum_t in [15:0]
        atomicAdd(&whist[2 * b], 0x10000u + (unsigned)t);
        // sum_p in fixed point, scale 2^16, round-to-nearest
        atomicAdd(&whist[2 * b + 1], __float2uint_rn(p * 65536.0f));
    }
}

__global__ void __launch_bounds__(TPB) ace_ws_zero(unsigned long long* ws) {
    if (threadIdx.x < 64) ws[threadIdx.x] = 0ull;
}

__global__ void __launch_bounds__(TPB) ace_hist_kernel(
        const float* __restrict__ preds, const int* __restrict__ tgts,
        int n, int tiles, unsigned long long* __restrict__ ws) {
    __shared__ float    s_p[2][TILE];
    __shared__ int      s_t[2][TILE];
    __shared__ unsigned s_hist[NREP * HSTRIDE];

    const int tid = threadIdx.x;
    unsigned* whist = &s_hist[(tid >> 4) * HSTRIDE];   // per-half-wave replica

    for (int i = tid; i < NREP * HSTRIDE; i += TPB) s_hist[i] = 0u;
    __syncthreads();

    const int G = gridDim.x;
    const unsigned lp0 = lds_off(&s_p[0][0]) + (unsigned)tid * 16u;
    const unsigned lp1 = lds_off(&s_p[1][0]) + (unsigned)tid * 16u;
    const unsigned lt0 = lds_off(&s_t[0][0]) + (unsigned)tid * 16u;
    const unsigned lt1 = lds_off(&s_t[1][0]) + (unsigned)tid * 16u;

    int tile = blockIdx.x;
    if (tile < tiles) {  // prologue: fill buffer 0 (uniform branch, EXEC full)
        unsigned long long gp = (unsigned long long)(uintptr_t)(preds + (size_t)tile * TILE) + (unsigned)tid * 16u;
        unsigned long long gt = (unsigned long long)(uintptr_t)(tgts  + (size_t)tile * TILE) + (unsigned)tid * 16u;
        async_cp16(lp0, gp); async_cp16(lp0 + 4096u, gp + 4096u);
        async_cp16(lt0, gt); async_cp16(lt0 + 4096u, gt + 4096u);
    }

    int buf = 0;
    for (; tile < tiles; tile += G) {
        const int nxt = tile + G;
        if (nxt < tiles) {
            // issue next tile into the *other* buffer, then wait for current;
            // ASYNCcnt completes in order, so <=4 outstanding => current done
            const unsigned lp = buf ? lp0 : lp1;
            const unsigned lt = buf ? lt0 : lt1;
            unsigned long long gp = (unsigned long long)(uintptr_t)(preds + (size_t)nxt * TILE) + (unsigned)tid * 16u;
            unsigned long long gt = (unsigned long long)(uintptr_t)(tgts  + (size_t)nxt * TILE) + (unsigned)tid * 16u;
            async_cp16(lp, gp); async_cp16(lp + 4096u, gp + 4096u);
            async_cp16(lt, gt); async_cp16(lt + 4096u, gt + 4096u);
            asm volatile("s_wait_asynccnt 0x4" ::: "memory");
        } else {
            asm volatile("s_wait_asynccnt 0x0" ::: "memory");
        }
        __syncthreads();   // all waves' async data for this tile visible

        const float4* pb = (const float4*)&s_p[buf][0];
        const int4*   tb = (const int4*)&s_t[buf][0];
#pragma unroll
        for (int j = 0; j < 2; j++) {
            float4 p = pb[tid + j * TPB];
            int4   t = tb[tid + j * TPB];
            acc_elem(p.x, t.x, whist);
            acc_elem(p.y, t.y, whist);
            acc_elem(p.z, t.z, whist);
            acc_elem(p.w, t.w, whist);
        }
        __syncthreads();   // buffer may be re-filled next iteration
        buf ^= 1;
    }

    // tail (n % TILE): block 0, direct guarded loads
    if (blockIdx.x == 0) {
        for (int i = tiles * TILE + tid; i < n; i += TPB)
            acc_elem(preds[i], tgts[i], whist);
    }
    __syncthreads();

    // reduce the 16 half-wave replicas, flush to global with integer atomics
    if (tid < N_BINS) {
        unsigned long long c = 0, st = 0, sp = 0;
        for (int r = 0; r < NREP; r++) {
            unsigned packed = s_hist[r * HSTRIDE + 2 * tid];
            unsigned sump   = s_hist[r * HSTRIDE + 2 * tid + 1];
            c  += packed >> 16;
            st += packed & 0xFFFFu;
            sp += sump;
        }
        atomicAdd(&ws[tid], c);
        atomicAdd(&ws[N_BINS + tid], st);
        atomicAdd(&ws[2 * N_BINS + tid], sp);
    }
}

__global__ void ace_finalize(const unsigned long long* __restrict__ ws,
                             float* __restrict__ out) {
    if (threadIdx.x == 0 && blockIdx.x == 0) {
        double ace = 0.0;
        for (int b = 0; b < N_BINS; b++) {
            unsigned long long c = ws[b];
            if (c != 0ull) {
                double inv = 1.0 / (double)c;
                double e = ((double)ws[2 * N_BINS + b] * (1.0 / 65536.0)) * inv;
                double o = (double)ws[N_BINS + b] * inv;
                ace += fabs(e - o);
            }
        }
        out[0] = (float)(ace / (double)N_BINS);
    }
}

extern "C" void kernel_launch(void* const* d_in, const int* in_sizes, int n_in,
                              void* d_out, int out_size, void* d_ws, size_t ws_size,
                              hipStream_t stream) {
    const float* preds = (const float*)d_in[0];
    const int*   tgts  = (const int*)d_in[1];
    float* out = (float*)d_out;
    unsigned long long* ws = (unsigned long long*)d_ws;
    const int n = in_sizes[0];

    const int tiles = n / TILE;
    int grid = tiles > MAX_GRID ? MAX_GRID : (tiles > 0 ? tiles : 1);
    if (tiles > grid * MAX_TILES_PER_BLOCK)
        grid = (tiles + MAX_TILES_PER_BLOCK - 1) / MAX_TILES_PER_BLOCK;

    ace_ws_zero<<<1, TPB, 0, stream>>>(ws);
    ace_hist_kernel<<<grid, TPB, 0, stream>>>(preds, tgts, n, tiles, ws);
    ace_finalize<<<1, 32, 0, stream>>>(ws, out);
}